// QpNet_335007449557
// MI455X (gfx1250) — compile-verified
//
#include <hip/hip_runtime.h>

// ---------------------------------------------------------------------------
// QP-MPC solver for MI455X (gfx1250).  B=8192 batch, nz=64, m=384, 30 IPM its.
// Core compute: M = Qhat + diag(d01) + Bhat^T diag(d23) Bhat via
// v_wmma_f32_16x16x4_f32 (f32 matrix pipe), Cholesky + solves in LDS.
// B_hat staged into LDS via global_load_async_to_lds_b128 (ASYNCcnt path).
// ---------------------------------------------------------------------------

#define NZ       64
#define MCON     384
#define BROWS    128
#define QP_ITERS 30
#define SIGMA    0.1f
#define BH_LD    72   // LDS row stride for Bhat: 2*72 % 64 == 16 -> half-wave
                      // K-row pairs of a WMMA fragment hit disjoint banks
#define M_LD     65   // LDS row stride for M: column access conflict-free

typedef __attribute__((ext_vector_type(2))) float v2f;
typedef __attribute__((ext_vector_type(8))) float v8f;

// workspace layout (float offsets)
#define WS_AHAT 0                     // 128*16
#define WS_BHAT (WS_AHAT + 128*16)    // 128*64
#define WS_QHAT (WS_BHAT + 128*64)    // 64*64
#define WS_QB   (WS_QHAT + 64*64)     // 128*64   (Q_diag @ B_hat)
#define WS_QM   (WS_QB   + 128*64)    // 16*16

// ---------------------------------------------------------------------------
// Kernel 0: build shared matrices (single block, trivial cost)
// ---------------------------------------------------------------------------
__global__ __launch_bounds__(256)
void qp_precompute(const float* __restrict__ Qp, const float* __restrict__ Rp,
                   const float* __restrict__ Ap, const float* __restrict__ Bp,
                   float* __restrict__ ws)
{
    __shared__ float sAp[256], sQp[256], sRp[64], sQm[256], sRm[64];
    __shared__ float cur[256];
    __shared__ float ab[128];
    __shared__ float sBh[128 * 64];

    const int tid = threadIdx.x;

    sAp[tid] = Ap[tid];
    sQp[tid] = Qp[tid];
    if (tid < 64)  sRp[tid] = Rp[tid];
    if (tid < 128) ab[tid]  = Bp[tid];
    for (int idx = tid; idx < 128 * 64; idx += 256) sBh[idx] = 0.0f;
    __syncthreads();

    // Qm = Qp Qp^T ; Rm = Rp Rp^T
    {
        int i = tid >> 4, j = tid & 15;
        float a = 0.0f;
        for (int k = 0; k < 16; ++k) a += sQp[i * 16 + k] * sQp[j * 16 + k];
        sQm[tid] = a;
        ws[WS_QM + tid] = a;
    }
    if (tid < 64) {
        int i = tid >> 3, j = tid & 7;
        float a = 0.0f;
        for (int k = 0; k < 8; ++k) a += sRp[i * 8 + k] * sRp[j * 8 + k];
        sRm[tid] = a;
    }
    cur[tid] = sAp[tid];
    __syncthreads();

    // A_hat = [A; A^2; ...; A^8]
    for (int t = 0; t < 8; ++t) {
        int i = tid >> 4, j = tid & 15;
        ws[WS_AHAT + (t * 16 + i) * 16 + j] = cur[tid];
        float a = 0.0f;
        for (int k = 0; k < 16; ++k) a += cur[i * 16 + k] * sAp[k * 16 + j];
        __syncthreads();
        cur[tid] = a;
        __syncthreads();
    }

    // B_hat block band: block (i, j) = A^(i-j) B for j <= i
    for (int t = 0; t < 8; ++t) {
        float a = 0.0f;
        if (tid < 128) {
            int r = tid >> 3, c = tid & 7;
            for (int i = t; i < 8; ++i)
                sBh[(i * 16 + r) * 64 + (i - t) * 8 + c] = ab[tid];
            for (int k = 0; k < 16; ++k) a += sAp[r * 16 + k] * ab[k * 8 + c];
        }
        __syncthreads();
        if (tid < 128) ab[tid] = a;
        __syncthreads();
    }

    for (int idx = tid; idx < 128 * 64; idx += 256) ws[WS_BHAT + idx] = sBh[idx];

    // QB = Q_diag @ B_hat  (block-diagonal Qm applied per 16-row block)
    for (int idx = tid; idx < 128 * 64; idx += 256) {
        int r = idx >> 6, c = idx & 63;
        float a = 0.0f;
        for (int k = 0; k < 16; ++k)
            a += sQm[(r & 15) * 16 + k] * sBh[((r >> 4) * 16 + k) * 64 + c];
        ws[WS_QB + idx] = a;
    }
    __syncthreads();

    // Q_hat = 2 * (B_hat^T QB + R_diag)
    for (int idx = tid; idx < 64 * 64; idx += 256) {
        int i = idx >> 6, j = idx & 63;
        float a = 0.0f;
        for (int r = 0; r < 128; ++r) a += sBh[r * 64 + i] * ws[WS_QB + r * 64 + j];
        if ((i >> 3) == (j >> 3)) a += sRm[(i & 7) * 8 + (j & 7)];
        ws[WS_QHAT + idx] = 2.0f * a;
    }
}

// ---------------------------------------------------------------------------
// Kernel 1: one block (8 wave32s) per batch element
// ---------------------------------------------------------------------------
__global__ __launch_bounds__(256)
void qp_solver(const float* __restrict__ x, const float* __restrict__ ws,
               float* __restrict__ out)
{
    const float* Ahat = ws + WS_AHAT;
    const float* Bhat = ws + WS_BHAT;
    const float* Qhat = ws + WS_QHAT;
    const float* QB   = ws + WS_QB;
    const float* Qm   = ws + WS_QM;

    __shared__ __align__(16) float sBh[128 * BH_LD];   // 36864 B
    __shared__ float sM[64 * M_LD];                    // 16640 B
    __shared__ float sS[MCON], sLam[MCON], sRp[MCON], sDs[MCON], sDl[MCON];
    __shared__ float sAx0[128], sBdz[128], sD23[128], sWd[128];
    __shared__ float sZ[64], sDz[64], sRhs[64], sP[64];
    __shared__ float sX0[16], sRed[8], sScal[2];

    const int tid  = threadIdx.x;
    const int lane = tid & 31;
    const int wave = tid >> 5;
    const int bid  = blockIdx.x;

    if (tid < 16) sX0[tid] = x[bid * 16 + tid];

    // --- stage B_hat into LDS via CDNA5 async LDS-DMA (ASYNCcnt tracked) ---
    // 128x64 floats as 2048 16-byte chunks; both global rows (256B) and LDS
    // rows (288B) are 16B aligned.  Low 32 bits of a generic pointer to
    // __shared__ == wave-relative LDS byte offset (flat shared aperture).
    for (int idx = tid; idx < 2048; idx += 256) {
        const int r  = idx >> 4;
        const int c4 = (idx & 15) * 4;
        unsigned lds_addr   = (unsigned)(size_t)&sBh[r * BH_LD + c4];
        const float* gaddr  = &Bhat[r * 64 + c4];
        asm volatile("global_load_async_to_lds_b128 %0, %1, off"
                     :: "v"(lds_addr), "v"(gaddr) : "memory");
    }

    // overlap: A_x0 = A_hat x0 does not need sBh
    __syncthreads();
    if (tid < 128) {
        float a = 0.0f;
        for (int c = 0; c < 16; ++c) a += Ahat[tid * 16 + c] * sX0[c];
        sAx0[tid] = a;
    }
    asm volatile("s_wait_asynccnt 0x0" ::: "memory");
    __syncthreads();

    if (tid < 64) {                               // p = 2 A_x0 QB ; z = 0
        float a = 0.0f;
        for (int r = 0; r < 128; ++r) a += sAx0[r] * QB[r * 64 + tid];
        sP[tid] = 2.0f * a;
        sZ[tid] = 0.0f;
    }
    for (int i = tid; i < MCON; i += 256) {       // s = max(h, 0.1); lam = 1
        float h = (i < 128) ? 10.0f
                : (i < 256) ? 4.0f - sAx0[i - 128]
                            : 4.0f + sAx0[i - 256];
        sS[i]   = fmaxf(h, 0.1f);
        sLam[i] = 1.0f;
    }
    __syncthreads();

    // ---------------------- interior point iterations ----------------------
    for (int it = 0; it < QP_ITERS; ++it) {
        // mu = SIGMA * sum(s*lam)/m
        float part = 0.0f;
        for (int i = tid; i < MCON; i += 256) part += sS[i] * sLam[i];
        #pragma unroll
        for (int off = 16; off > 0; off >>= 1) part += __shfl_down(part, off, 32);
        if (lane == 0) sRed[wave] = part;
        __syncthreads();
        if (tid == 0) {
            float a = 0.0f;
            for (int w = 0; w < 8; ++w) a += sRed[w];
            sScal[0] = SIGMA * a / (float)MCON;
        }
        __syncthreads();
        const float mu = sScal[0];

        // Bz (reuse sBdz)
        if (tid < 128) {
            float a = 0.0f;
            for (int c = 0; c < 64; ++c) a += sBh[tid * BH_LD + c] * sZ[c];
            sBdz[tid] = a;
        }
        __syncthreads();

        // r_prim = G z + s - h
        for (int i = tid; i < MCON; i += 256) {
            float gz = (i < 64)  ?  sZ[i]
                     : (i < 128) ? -sZ[i - 64]
                     : (i < 256) ?  sBdz[i - 128]
                                 : -sBdz[i - 256];
            float h  = (i < 128) ? 10.0f
                     : (i < 256) ? 4.0f - sAx0[i - 128]
                                 : 4.0f + sAx0[i - 256];
            sRp[i] = gz + sS[i] - h;
        }
        __syncthreads();

        // d23 (WMMA scaling) and wd = w2 - w3 (for rhs)
        if (tid < 128) {
            float d2 = sLam[128 + tid] / sS[128 + tid];
            float d3 = sLam[256 + tid] / sS[256 + tid];
            sD23[tid] = d2 + d3;
            float w2 = mu / sS[128 + tid] - sLam[128 + tid] + d2 * sRp[128 + tid];
            float w3 = mu / sS[256 + tid] - sLam[256 + tid] + d3 * sRp[256 + tid];
            sWd[tid] = w2 - w3;
        }
        __syncthreads();

        // ---- M = Qhat + diag(d01) + Bhat^T diag(d23) Bhat via f32 WMMA ----
        // 16 tiles of 16x16; each wave owns 2 tiles sharing the A fragment.
        {
            const int ti   = wave >> 1;
            const int tjA  = (wave & 1) * 2;
            const int half = lane >> 4;
            const int n    = lane & 15;
            const int ci   = ti * 16 + n;
            const int cj0  = tjA * 16 + n;
            const int cj1  = (tjA + 1) * 16 + n;
            v8f acc0 = {};
            v8f acc1 = {};
            for (int kb = 0; kb < 32; ++kb) {
                const int k0 = kb * 4 + 2 * half;     // K = vgpr + 2*half
                v2f a, b0, b1;
                a[0] = sBh[k0 * BH_LD + ci];
                a[1] = sBh[(k0 + 1) * BH_LD + ci];
                const float dA = sD23[k0], dB = sD23[k0 + 1];
                b0[0] = dA * sBh[k0 * BH_LD + cj0];
                b0[1] = dB * sBh[(k0 + 1) * BH_LD + cj0];
                b1[0] = dA * sBh[k0 * BH_LD + cj1];
                b1[1] = dB * sBh[(k0 + 1) * BH_LD + cj1];
                acc0 = __builtin_amdgcn_wmma_f32_16x16x4_f32(
                    false, a, false, b0, (short)0, acc0, false, false);
                acc1 = __builtin_amdgcn_wmma_f32_16x16x4_f32(
                    false, a, false, b1, (short)0, acc1, false, false);
            }
            #pragma unroll
            for (int r = 0; r < 8; ++r) {
                const int gm = ti * 16 + r + 8 * half;
                const float dg = sLam[gm] / sS[gm] + sLam[64 + gm] / sS[64 + gm];
                const int gn0 = tjA * 16 + n;
                float v0 = acc0[r] + Qhat[gm * 64 + gn0] + ((gm == gn0) ? dg : 0.0f);
                sM[gm * M_LD + gn0] = v0;
                const int gn1 = (tjA + 1) * 16 + n;
                float v1 = acc1[r] + Qhat[gm * 64 + gn1] + ((gm == gn1) ? dg : 0.0f);
                sM[gm * M_LD + gn1] = v1;
            }
        }

        // rhs = -(r_dual + (mu/s - lam + d*r_prim) @ G)
        if (tid < 64) {
            float rd = sP[tid] + sLam[tid] - sLam[64 + tid];
            for (int j = 0; j < 64; ++j) rd += Qhat[tid * 64 + j] * sZ[j];
            float bt = 0.0f, btw = 0.0f;
            for (int r = 0; r < 128; ++r) {
                const float bc = sBh[r * BH_LD + tid];
                bt  += bc * (sLam[128 + r] - sLam[256 + r]);
                btw += bc * sWd[r];
            }
            rd += bt;
            const float d0 = sLam[tid] / sS[tid];
            const float d1 = sLam[64 + tid] / sS[64 + tid];
            const float w0 = mu / sS[tid] - sLam[tid] + d0 * sRp[tid];
            const float w1 = mu / sS[64 + tid] - sLam[64 + tid] + d1 * sRp[64 + tid];
            sRhs[tid] = -(rd + (w0 - w1) + btw);
        }
        __syncthreads();

        // ---- in-place Cholesky of M (lower); trailing update uses all
        //      256 threads: 4 threads per row, columns strided by 4 ----
        for (int k = 0; k < 64; ++k) {
            if (tid == 0) sM[k * M_LD + k] = sqrtf(sM[k * M_LD + k]);
            __syncthreads();
            const float dk = sM[k * M_LD + k];
            if (tid > k && tid < 64) sM[tid * M_LD + k] /= dk;
            __syncthreads();
            {
                const int i = tid & 63;
                const int q = tid >> 6;
                if (i > k) {
                    const float lik = sM[i * M_LD + k];
                    for (int j = k + 1 + q; j <= i; j += 4)
                        sM[i * M_LD + j] -= lik * sM[j * M_LD + k];
                }
            }
            __syncthreads();
        }

        // forward solve L y = rhs (in place)
        for (int k = 0; k < 64; ++k) {
            if (tid == 0) sRhs[k] /= sM[k * M_LD + k];
            __syncthreads();
            const float yk = sRhs[k];
            if (tid > k && tid < 64) sRhs[tid] -= sM[tid * M_LD + k] * yk;
            __syncthreads();
        }
        // backward solve L^T dz = y (in place)
        for (int k = 63; k >= 0; --k) {
            if (tid == 0) sRhs[k] /= sM[k * M_LD + k];
            __syncthreads();
            const float zk = sRhs[k];
            if (tid < k) sRhs[tid] -= sM[k * M_LD + tid] * zk;
            __syncthreads();
        }
        if (tid < 64) sDz[tid] = sRhs[tid];
        __syncthreads();

        // B dz
        if (tid < 128) {
            float a = 0.0f;
            for (int c = 0; c < 64; ++c) a += sBh[tid * BH_LD + c] * sDz[c];
            sBdz[tid] = a;
        }
        __syncthreads();

        // ds, dlam
        for (int i = tid; i < MCON; i += 256) {
            const float gdz = (i < 64)  ?  sDz[i]
                            : (i < 128) ? -sDz[i - 64]
                            : (i < 256) ?  sBdz[i - 128]
                                        : -sBdz[i - 256];
            const float di = sLam[i] / sS[i];
            sDs[i] = -sRp[i] - gdz;
            sDl[i] = mu / sS[i] - sLam[i] + di * (sRp[i] + gdz);
        }
        __syncthreads();

        // fraction-to-boundary step
        float amin = 3.0e38f;
        for (int i = tid; i < MCON; i += 256) {
            const float dsi = sDs[i];
            if (dsi < 0.0f) amin = fminf(amin, -sS[i] / dsi);
            const float dli = sDl[i];
            if (dli < 0.0f) amin = fminf(amin, -sLam[i] / dli);
        }
        #pragma unroll
        for (int off = 16; off > 0; off >>= 1)
            amin = fminf(amin, __shfl_down(amin, off, 32));
        if (lane == 0) sRed[wave] = amin;
        __syncthreads();
        if (tid == 0) {
            float a = 3.0e38f;
            for (int w = 0; w < 8; ++w) a = fminf(a, sRed[w]);
            sScal[1] = fminf(1.0f, 0.99f * a);
        }
        __syncthreads();
        const float alpha = sScal[1];

        if (tid < 64) sZ[tid] += alpha * sDz[tid];
        for (int i = tid; i < MCON; i += 256) {
            sS[i]   += alpha * sDs[i];
            sLam[i] += alpha * sDl[i];
        }
        __syncthreads();
    }

    // ------------------------------ outputs -------------------------------
    float partc = 0.0f;
    if (tid < 64) {                     // a = sum(u Qh u / 2 + p u)
        float q = 0.0f;
        for (int j = 0; j < 64; ++j) q += Qhat[tid * 64 + j] * sZ[j];
        partc += (0.5f * q + sP[tid]) * sZ[tid];
    }
    if (tid < 128) {                    // b = A_x0 Q_diag A_x0
        const int blk = tid >> 4, cc = tid & 15;
        float t = 0.0f;
        for (int j = 0; j < 16; ++j) t += sAx0[blk * 16 + j] * Qm[j * 16 + cc];
        partc += t * sAx0[tid];
    }
    if (tid < 16) {                     // c = x0 Qm x0
        float t = 0.0f;
        for (int j = 0; j < 16; ++j) t += Qm[tid * 16 + j] * sX0[j];
        partc += sX0[tid] * t;
    }
    #pragma unroll
    for (int off = 16; off > 0; off >>= 1) partc += __shfl_down(partc, off, 32);
    if (lane == 0) sRed[wave] = partc;
    __syncthreads();
    if (tid == 0) {
        float a = 0.0f;
        for (int w = 0; w < 8; ++w) a += sRed[w];
        out[bid * 65] = 0.1f * a;
    }
    if (tid < 64) out[bid * 65 + 1 + tid] = sZ[tid];
}

// ---------------------------------------------------------------------------
extern "C" void kernel_launch(void* const* d_in, const int* in_sizes, int n_in,
                              void* d_out, int out_size, void* d_ws, size_t ws_size,
                              hipStream_t stream)
{
    const float* x  = (const float*)d_in[0];
    const float* Qp = (const float*)d_in[1];
    const float* Rp = (const float*)d_in[2];
    const float* Ap = (const float*)d_in[3];
    const float* Bp = (const float*)d_in[4];
    float* ws  = (float*)d_ws;
    float* outp = (float*)d_out;
    const int B = in_sizes[0] / 16;

    qp_precompute<<<1, 256, 0, stream>>>(Qp, Rp, Ap, Bp, ws);
    qp_solver<<<B, 256, 0, stream>>>(x, ws, outp);
}